// SpaceGroupDenseLayer_78494822302097
// MI455X (gfx1250) — compile-verified
//
#include <hip/hip_runtime.h>

// ---- CDNA5 WMMA vector types ----
typedef __attribute__((ext_vector_type(2))) float v2f;
typedef __attribute__((ext_vector_type(8))) float v8f;

#define BATCH 64
#define MDIM  256   // rows per batch (N in the reference)
#define KDIM  256   // INPUT_DIM
#define FDIM  512   // FEATURES

#define FB 64           // F columns per workgroup
#define KC 32           // K chunk staged in LDS (16 K-pairs)
#define PSTRIDE 80      // pair-row stride in float2 units: 160 floats ≡ 32 mod 64
                        // -> ds_load_b64 fragment reads are bank-conflict-free

__global__ __launch_bounds__(256)
void sg_dense_wmma(const float* __restrict__ xr, const float* __restrict__ xi,
                   const int*   __restrict__ sg,
                   const float* __restrict__ wr, const float* __restrict__ wi,
                   const float* __restrict__ brl, const float* __restrict__ bim,
                   float* __restrict__ out)
{
    // K-pair-interleaved weight tiles: float2 {W[2p][n], W[2p+1][n]} at
    // float index (p*PSTRIDE + n)*2.  16 pairs x 80 float2 = 2560 floats.
    __shared__ float lwr[(KC / 2) * PSTRIDE * 2];
    __shared__ float lwi[(KC / 2) * PSTRIDE * 2];

    const int b    = blockIdx.z;
    const int fb   = blockIdx.x * FB;
    const int tid  = threadIdx.x;
    const int wave = tid >> 5;
    const int lane = tid & 31;
    const int lrow  = lane & 15;   // M (A-frag) or N (B/C-frag) within a tile
    const int pairh = lane >> 4;   // upper half-wave reads K-pair +1 (K+2,K+3)
    const int khalf = pairh << 1;

    const int sgv = sg[b];
    const float* wrp = wr + (size_t)sgv * (KDIM * FDIM) + fb;
    const float* wip = wi + (size_t)sgv * (KDIM * FDIM) + fb;

    const int m0 = wave * 32;      // this wave's 32-row M strip
    const float* xr0 = xr + ((size_t)b * MDIM + m0 + lrow) * KDIM;
    const float* xi0 = xi + ((size_t)b * MDIM + m0 + lrow) * KDIM;
    const float* xr1 = xr0 + (size_t)16 * KDIM;
    const float* xi1 = xi0 + (size_t)16 * KDIM;

    // Staging assignment: 32 column-pairs x 16 K-pairs per buffer.
    const int np  = tid & 31;      // column pair -> cols 2np, 2np+1
    const int pr0 = tid >> 5;      // K-pair rows 0..7 (+8 on second pass)

    v8f accR[2][4];
    v8f accI[2][4];
    #pragma unroll
    for (int t = 0; t < 2; ++t)
        #pragma unroll
        for (int f = 0; f < 4; ++f) {
            accR[t][f] = (v8f)0.0f;
            accI[t][f] = (v8f)0.0f;
        }

    for (int kc = 0; kc < KDIM; kc += KC) {
        __syncthreads();
        // Cooperative stage of Wr/Wi chunk (KC x FB) into pair-interleaved LDS.
        #pragma unroll
        for (int r = 0; r < 2; ++r) {
            const int pair = pr0 + r * 8;             // 0..15
            const size_t g = (size_t)(kc + pair * 2) * FDIM + np * 2;
            const v2f ar = *(const v2f*)(wrp + g);            // row k,   cols n,n+1
            const v2f cr = *(const v2f*)(wrp + g + FDIM);     // row k+1, cols n,n+1
            const v2f ai = *(const v2f*)(wip + g);
            const v2f ci = *(const v2f*)(wip + g + FDIM);
            float4 sr; sr.x = ar.x; sr.y = cr.x; sr.z = ar.y; sr.w = cr.y;
            float4 si; si.x = ai.x; si.y = ci.x; si.z = ai.y; si.w = ci.y;
            const int li = (pair * PSTRIDE + np * 2) * 2;     // float index
            *(float4*)(&lwr[li]) = sr;
            *(float4*)(&lwi[li]) = si;
            if (kc + KC < KDIM) {   // uniform guard: prefetch next K-chunk
                __builtin_prefetch(wrp + g + (size_t)KC * FDIM, 0, 1);
                __builtin_prefetch(wip + g + (size_t)KC * FDIM, 0, 1);
            }
        }
        __syncthreads();

        #pragma unroll
        for (int k4 = 0; k4 < KC; k4 += 4) {
            const int kg = kc + k4 + khalf;
            // A fragments: 16x4 f32, lane<16 -> K {0,1}, lane>=16 -> K {2,3}
            v2f ar0 = *(const v2f*)(xr0 + kg);
            v2f ai0 = *(const v2f*)(xi0 + kg);
            v2f ar1 = *(const v2f*)(xr1 + kg);
            v2f ai1 = *(const v2f*)(xi1 + kg);
            v2f nai0 = -ai0;   // f32 WMMA has no A/B negate modifier
            v2f nai1 = -ai1;

            const int pbase = (k4 >> 1) + pairh;      // K-pair for this half-wave
            #pragma unroll
            for (int f = 0; f < 4; ++f) {
                const int n = f * 16 + lrow;
                // Single b64 per fragment, lands in an even-aligned VGPR pair.
                const v2f bfr = *(const v2f*)(&lwr[(pbase * PSTRIDE + n) * 2]);
                const v2f bfi = *(const v2f*)(&lwi[(pbase * PSTRIDE + n) * 2]);

                // Cr += Ar*Br + (-Ai)*Bi ; Ci += Ar*Bi + Ai*Br
                accR[0][f] = __builtin_amdgcn_wmma_f32_16x16x4_f32(
                    false, ar0,  false, bfr, (short)0, accR[0][f], false, false);
                accR[0][f] = __builtin_amdgcn_wmma_f32_16x16x4_f32(
                    false, nai0, false, bfi, (short)0, accR[0][f], false, false);
                accI[0][f] = __builtin_amdgcn_wmma_f32_16x16x4_f32(
                    false, ar0,  false, bfi, (short)0, accI[0][f], false, false);
                accI[0][f] = __builtin_amdgcn_wmma_f32_16x16x4_f32(
                    false, ai0,  false, bfr, (short)0, accI[0][f], false, false);

                accR[1][f] = __builtin_amdgcn_wmma_f32_16x16x4_f32(
                    false, ar1,  false, bfr, (short)0, accR[1][f], false, false);
                accR[1][f] = __builtin_amdgcn_wmma_f32_16x16x4_f32(
                    false, nai1, false, bfi, (short)0, accR[1][f], false, false);
                accI[1][f] = __builtin_amdgcn_wmma_f32_16x16x4_f32(
                    false, ar1,  false, bfi, (short)0, accI[1][f], false, false);
                accI[1][f] = __builtin_amdgcn_wmma_f32_16x16x4_f32(
                    false, ai1,  false, bfr, (short)0, accI[1][f], false, false);
            }
        }
    }

    // Bias add + interleaved complex store.
    // C/D layout: VGPR v, lane<16 -> M=v, N=lane ; lane>=16 -> M=v+8, N=lane-16
    #pragma unroll
    for (int f = 0; f < 4; ++f) {
        const int col = fb + f * 16 + lrow;
        const float bre = brl[(size_t)sgv * FDIM + col];
        const float bie = bim[(size_t)sgv * FDIM + col];
        #pragma unroll
        for (int t = 0; t < 2; ++t) {
            const int mb = m0 + t * 16 + (pairh << 3);
            #pragma unroll
            for (int v = 0; v < 8; ++v) {
                const int M = mb + v;
                float2 o;
                o.x = accR[t][f][v] + bre;
                o.y = accI[t][f][v] + bie;
                *(float2*)(out + (((size_t)b * MDIM + M) * FDIM + col) * 2) = o;
            }
        }
    }
}

extern "C" void kernel_launch(void* const* d_in, const int* in_sizes, int n_in,
                              void* d_out, int out_size, void* d_ws, size_t ws_size,
                              hipStream_t stream) {
    (void)in_sizes; (void)n_in; (void)out_size; (void)d_ws; (void)ws_size;
    const float* xr  = (const float*)d_in[0];
    const float* xi  = (const float*)d_in[1];
    const int*   sg  = (const int*)  d_in[2];
    const float* wr  = (const float*)d_in[3];
    const float* wi  = (const float*)d_in[4];
    const float* brl = (const float*)d_in[5];
    const float* bim = (const float*)d_in[6];
    float* out = (float*)d_out;

    dim3 grid(FDIM / FB, 1, BATCH);   // (8, 1, 64)
    dim3 block(256, 1, 1);            // 8 wave32s
    sg_dense_wmma<<<grid, block, 0, stream>>>(xr, xi, sg, wr, wi, brl, bim, out);
}